// SeqSelfAttention_29875792511793
// MI455X (gfx1250) — compile-verified
//
#include <hip/hip_runtime.h>
#include <hip/hip_bf16.h>
#include <math.h>

#define BB 4
#define LL 1024
#define DD 64
#define UU 32

#if defined(__has_builtin)
#  if __has_builtin(__builtin_amdgcn_tanhf)
#    define TANHF(x) __builtin_amdgcn_tanhf(x)
#  else
#    define TANHF(x) tanhf(x)
#  endif
#else
#  define TANHF(x) tanhf(x)
#endif

#if defined(__has_builtin)
#  if __has_builtin(__builtin_amdgcn_s_wait_asynccnt)
#    define WAIT_ASYNC(n) __builtin_amdgcn_s_wait_asynccnt(n)
#  else
#    define WAIT_ASYNC(n) asm volatile("s_wait_asynccnt %0" :: "i"(n) : "memory")
#  endif
#else
#  define WAIT_ASYNC(n) asm volatile("s_wait_asynccnt %0" :: "i"(n) : "memory")
#endif

typedef __attribute__((ext_vector_type(2))) float v2f;
typedef __attribute__((ext_vector_type(8))) float v8f;

// One global_load_async_to_lds_b128: per-lane 16B, memory -> LDS, ASYNCcnt++.
// Generic pointer to __shared__ has the LDS byte address in its low 32 bits
// (aperture mapping).
__device__ __forceinline__ void async_b128(void* lds_ptr, const void* gptr) {
    const unsigned int lds_addr = (unsigned int)(uintptr_t)lds_ptr;
    asm volatile("global_load_async_to_lds_b128 %0, %1, off"
                 :: "v"(lds_addr), "v"(gptr) : "memory");
}

// ---------------------------------------------------------------------------
// Phase 1: q' = X @ Wt + bh,  k = X @ Wx.   One wave (32 lanes) per row.
// ---------------------------------------------------------------------------
__global__ __launch_bounds__(256)
void qk_proj_kernel(const float* __restrict__ x,
                    const float* __restrict__ Wt,
                    const float* __restrict__ Wx,
                    const float* __restrict__ bh,
                    float* __restrict__ qbuf,
                    float* __restrict__ kbuf) {
    const int lane = threadIdx.x & 31;
    const int wave = threadIdx.x >> 5;
    const int row  = blockIdx.x * 8 + wave;          // 0 .. B*L-1
    if (row >= BB * LL) return;

    const float* xr = x + (size_t)row * DD;
    const float x0 = xr[lane];
    const float x1 = xr[lane + 32];

    float qa = 0.f, ka = 0.f;
#pragma unroll
    for (int d = 0; d < 32; ++d) {
        const float xb = __shfl(x0, d, 32);
        qa = fmaf(xb, Wt[d * UU + lane], qa);
        ka = fmaf(xb, Wx[d * UU + lane], ka);
    }
#pragma unroll
    for (int d = 0; d < 32; ++d) {
        const float xb = __shfl(x1, d, 32);
        qa = fmaf(xb, Wt[(d + 32) * UU + lane], qa);
        ka = fmaf(xb, Wx[(d + 32) * UU + lane], ka);
    }
    qbuf[(size_t)row * UU + lane] = qa + bh[lane];   // fold bh into q
    kbuf[(size_t)row * UU + lane] = ka;
}

// ---------------------------------------------------------------------------
// Phase 2: fused flash attention, double-buffered async LDS staging.
//   e[i,j] = sum_u Wa[u] * tanh(q'[i,u] + k[j,u])      (ba cancels in softmax)
//   online softmax over j; acc[i,:] += P * V via V_WMMA_F32_16X16X4_F32.
// Block = 128 threads = 4 waves; each wave owns one 16-row i-tile; the block
// shares the per-j-tile k (16x32) and V (16x64) tiles, staged with
// GLOBAL_LOAD_ASYNC_TO_LDS_B128 one tile ahead of compute.
//
// WMMA fp32 fragment layouts (ISA 7.12.2, wave32):
//   A 16x4 : lanes 0-15 -> M=lane, K={0,1} in vgpr{0,1}; lanes 16-31 K={2,3}
//   B 4x16 : vgpr0 -> K=0(lo half)/K=2(hi half), vgpr1 -> K=1/K=3; N=lane&15
//   C 16x16: vgpr r -> M=r (lanes 0-15) / M=r+8 (lanes 16-31); N=lane&15
// ---------------------------------------------------------------------------
__global__ __launch_bounds__(128)
void attn_fused_kernel(const float* __restrict__ x,
                       const float* __restrict__ Wa,
                       const float* __restrict__ qbuf,
                       const float* __restrict__ kbuf,
                       float* __restrict__ out) {
    __shared__ __align__(16) float k_lds[2][16 * UU];   // 2 x 2 KB
    __shared__ __align__(16) float v_lds[2][16 * DD];   // 2 x 4 KB

    const int tid  = threadIdx.x;
    const int lane = tid & 31;
    const int wave = tid >> 5;
    const int half = lane >> 4;        // 0 | 1
    const int irow = lane & 15;        // A-fragment row

    const int blocksPerBatch = LL / 64;                 // 16
    const int b      = blockIdx.x / blocksPerBatch;
    const int i_base = (blockIdx.x % blocksPerBatch) * 64 + wave * 16;

    // per-lane copies of q row (row = i_base + irow, duplicated across halves)
    float q[UU], wa[UU];
    {
        const float* qr = qbuf + (size_t)(b * LL + i_base + irow) * UU;
#pragma unroll
        for (int u = 0; u < UU; ++u) { q[u] = qr[u]; wa[u] = Wa[u]; }
    }

    // 3 async b128 per thread per tile: 2 KB k-tile + 4 KB V-tile per block
    auto issue_tile = [&](int jt, int buf) {
        const float4* ks = (const float4*)(kbuf + (size_t)(b * LL + jt) * UU);
        async_b128(&k_lds[buf][tid * 4], ks + tid);
        const float4* vs = (const float4*)(x + (size_t)(b * LL + jt) * DD);
        async_b128(&v_lds[buf][tid * 4],         vs + tid);
        async_b128(&v_lds[buf][(tid + 128) * 4], vs + tid + 128);
    };

    float m = -__builtin_inff();
    float s = 0.f;
    v8f acc[4] = {};                   // 4 D-chunks of 16 cols each

    issue_tile(0, 0);                  // prologue prefetch

    for (int jt = 0; jt < LL; jt += 16) {
        const int cur = (jt >> 4) & 1;
        if (jt + 16 < LL) {
            issue_tile(jt + 16, cur ^ 1);   // prefetch next tile
            WAIT_ASYNC(3);                  // in-order: tile jt's 3 ops done
        } else {
            WAIT_ASYNC(0);
        }
        __syncthreads();               // tile jt visible to all waves

        const float* kt = k_lds[cur];
        const float* vt = v_lds[cur];

        // --- scores: 8 (i,j) pairs per lane, in A-fragment order
        float p[8];
        float tmax = -__builtin_inff();
#pragma unroll
        for (int c = 0; c < 4; ++c) {
#pragma unroll
            for (int t = 0; t < 2; ++t) {
                const int jj = 4 * c + 2 * half + t;
                const float4* krow = (const float4*)(kt + jj * UU);
                float e = 0.f;
#pragma unroll
                for (int u4 = 0; u4 < 8; ++u4) {
                    const float4 kv = krow[u4];
                    const int u = u4 * 4;
                    e = fmaf(wa[u + 0], TANHF(q[u + 0] + kv.x), e);
                    e = fmaf(wa[u + 1], TANHF(q[u + 1] + kv.y), e);
                    e = fmaf(wa[u + 2], TANHF(q[u + 2] + kv.z), e);
                    e = fmaf(wa[u + 3], TANHF(q[u + 3] + kv.w), e);
                }
                p[c * 2 + t] = e;
                tmax = fmaxf(tmax, e);
            }
        }
        // row max lives split across lane and lane^16
        tmax = fmaxf(tmax, __shfl_xor(tmax, 16, 32));
        const float mnew  = fmaxf(m, tmax);
        const float scale = __expf(m - mnew);        // 0 on first tile
        float psum = 0.f;
#pragma unroll
        for (int z = 0; z < 8; ++z) { p[z] = __expf(p[z] - mnew); psum += p[z]; }
        psum += __shfl_xor(psum, 16, 32);
        s = s * scale + psum;
        m = mnew;

        // rescale accumulator rows only when some row's max moved
        if (__ballot(scale != 1.0f)) {
#pragma unroll
            for (int r = 0; r < 8; ++r) {
                const float rs = __shfl(scale, r + (half << 3), 32);
                acc[0][r] *= rs; acc[1][r] *= rs;
                acc[2][r] *= rs; acc[3][r] *= rs;
            }
        }

        // --- P (16x16) @ V (16x64) via 4x4 fp32 WMMA 16x16x4
#pragma unroll
        for (int c = 0; c < 4; ++c) {
            v2f a; a.x = p[2 * c]; a.y = p[2 * c + 1];
            const int jb = 4 * c + 2 * half;         // B-fragment K rows
#pragma unroll
            for (int n = 0; n < 4; ++n) {
                v2f bf;
                bf.x = vt[jb * DD       + n * 16 + irow];
                bf.y = vt[(jb + 1) * DD + n * 16 + irow];
                acc[n] = __builtin_amdgcn_wmma_f32_16x16x4_f32(
                    /*neg_a=*/false, a, /*neg_b=*/false, bf,
                    /*c_mod=*/(short)0, acc[n],
                    /*reuse_a=*/false, /*reuse_b=*/false);
            }
        }
        __syncthreads();               // done reading buf before t+2 overwrite
    }

    // --- epilogue: v = acc / (s_row + 1e-8)   (matches reference denominator)
#pragma unroll
    for (int r = 0; r < 8; ++r) {
        const float sr  = __shfl(s, r + (half << 3), 32);
        const float inv = 1.0f / (sr + 1e-8f);
        const int   row = r + 8 * half;
        float* orow = out + (size_t)(b * LL + i_base + row) * DD;
#pragma unroll
        for (int n = 0; n < 4; ++n)
            orow[n * 16 + irow] = acc[n][r] * inv;
    }
}

// ---------------------------------------------------------------------------
extern "C" void kernel_launch(void* const* d_in, const int* in_sizes, int n_in,
                              void* d_out, int out_size, void* d_ws, size_t ws_size,
                              hipStream_t stream) {
    const float* inputs = (const float*)d_in[0];   // (B,L,D)
    const float* Wt     = (const float*)d_in[1];   // (D,U)
    const float* Wx     = (const float*)d_in[2];   // (D,U)
    const float* Wa     = (const float*)d_in[3];   // (U,1)
    const float* bh     = (const float*)d_in[4];   // (U,)
    // d_in[5] = ba   : constant shift of all logits -> cancels in softmax
    // d_in[6] = attention_width : dead code in reference
    float* out = (float*)d_out;                    // (B,L,D)

    float* qbuf = (float*)d_ws;                    // B*L*U floats
    float* kbuf = qbuf + (size_t)BB * LL * UU;     // B*L*U floats (1 MB total)

    qk_proj_kernel<<<(BB * LL) / 8, 256, 0, stream>>>(inputs, Wt, Wx, bh, qbuf, kbuf);
    attn_fused_kernel<<<BB * (LL / 64), 128, 0, stream>>>(inputs, Wa, qbuf, kbuf, out);
}